// MultiHeadAttention_84473416778245
// MI455X (gfx1250) — compile-verified
//
#include <hip/hip_runtime.h>

// MI455X / gfx1250, wave32. fp32 attention via V_WMMA_F32_16X16X4_F32.
// B=8, S=2048, D=384, H=8, hd=48.  ~71 GFLOP fp32, all operands L2-resident
// (x=25MB, weights<2MB, L2=192MB) -> matrix-core bound; softmax is over the
// HEAD axis (8 values, local per (q,k)) so scores are never materialized.
// Projection GEMMs use 16x64 tiles per wave: A fragment reused 4x -> 1.25
// b64 loads per WMMA instead of 2 (load-path roofline fix).

typedef __attribute__((ext_vector_type(2))) float v2f;
typedef __attribute__((ext_vector_type(8))) float v8f;

#define D_MODEL  384
#define THREE_D  1152
#define N_HEADS  8
#define HEAD_DIM 48
#define BATCH    8
#define SEQ      2048
#define M_TOTAL  (BATCH * SEQ)   // 16384

static __device__ __forceinline__ v2f mkv2(float x, float y) {
  v2f r; r.x = x; r.y = y; return r;
}

// D = A(16x4 f32) * B(4x16 f32) + C(16x16 f32)
static __device__ __forceinline__ v8f wmma_f32(v2f a, v2f b, v8f c) {
  return __builtin_amdgcn_wmma_f32_16x16x4_f32(
      /*neg_a=*/false, a, /*neg_b=*/false, b,
      /*c_mod=*/(short)0, c, /*reuse_a=*/false, /*reuse_b=*/false);
}

// ---------------------------------------------------------------------------
// Kernel A: qkv[m,e] = x[m,:] . W_qkv[e,:] + b_qkv[e]; 16x64 tile per wave.
// Scatter into Q[b,h,s,d], K[b,h,s,d] (d contiguous), Vt[b,h,d,s] (s contig).
// e -> h = e/144, r = e%144 : r<48 Q, r<96 K, else V.
// ---------------------------------------------------------------------------
__global__ __launch_bounds__(256) void qkv_kernel(
    const float* __restrict__ x, const float* __restrict__ Wqkv,
    const float* __restrict__ bqkv,
    float* __restrict__ Q, float* __restrict__ K, float* __restrict__ Vt) {
  const int lane  = threadIdx.x;
  const int lmod  = lane & 15;
  const int lhalf = lane >> 4;
  const int tile  = blockIdx.x * 8 + threadIdx.y;      // 16x64 tiles
  const int mbase = (tile / (THREE_D / 64)) * 16;
  const int ebase = (tile % (THREE_D / 64)) * 64;

  const float* arow = x + (size_t)(mbase + lmod) * D_MODEL + 2 * lhalf;
  const float* brow0 = Wqkv + (size_t)(ebase +  0 + lmod) * D_MODEL + 2 * lhalf;
  const float* brow1 = Wqkv + (size_t)(ebase + 16 + lmod) * D_MODEL + 2 * lhalf;
  const float* brow2 = Wqkv + (size_t)(ebase + 32 + lmod) * D_MODEL + 2 * lhalf;
  const float* brow3 = Wqkv + (size_t)(ebase + 48 + lmod) * D_MODEL + 2 * lhalf;

  v8f acc0 = {}, acc1 = {}, acc2 = {}, acc3 = {};
#pragma unroll 4
  for (int c = 0; c < D_MODEL / 4; ++c) {
    const float2 af = *reinterpret_cast<const float2*>(arow  + 4 * c);
    const float2 b0 = *reinterpret_cast<const float2*>(brow0 + 4 * c);
    const float2 b1 = *reinterpret_cast<const float2*>(brow1 + 4 * c);
    const float2 b2 = *reinterpret_cast<const float2*>(brow2 + 4 * c);
    const float2 b3 = *reinterpret_cast<const float2*>(brow3 + 4 * c);
    const v2f av = mkv2(af.x, af.y);
    acc0 = wmma_f32(av, mkv2(b0.x, b0.y), acc0);
    acc1 = wmma_f32(av, mkv2(b1.x, b1.y), acc1);
    acc2 = wmma_f32(av, mkv2(b2.x, b2.y), acc2);
    acc3 = wmma_f32(av, mkv2(b3.x, b3.y), acc3);
  }

  v8f accs[4] = {acc0, acc1, acc2, acc3};
#pragma unroll
  for (int j = 0; j < 4; ++j) {
    const int   e    = ebase + 16 * j + lmod;
    const float bias = bqkv[e];
    const int   h    = e / 144;
    const int   rr   = e % 144;
    const size_t bhoff = (size_t)h;   // combined below
#pragma unroll
    for (int r = 0; r < 8; ++r) {
      const int m = mbase + r + 8 * lhalf;
      const int b = m >> 11;          // /2048
      const int s = m & 2047;
      const float val = accs[j][r] + bias;
      const size_t bh = (size_t)(b * N_HEADS) + bhoff;
      if (rr < 48) {
        Q[(bh * SEQ + s) * HEAD_DIM + rr] = val;
      } else if (rr < 96) {
        K[(bh * SEQ + s) * HEAD_DIM + (rr - 48)] = val;
      } else {
        Vt[(bh * HEAD_DIM + (rr - 96)) * SEQ + s] = val;
      }
    }
  }
}

// ---------------------------------------------------------------------------
// Kernel B: streaming attention. Block = (32,8): wave h owns head h for one
// 16-row q-tile. Per k-tile: scores via 12 wmma -> LDS -> 8-way cross-head
// softmax (wave h handles D-layout row r==h, covering all 256 (q,k) once)
// -> reload as A-fragments (transpose via LDS) -> 12 wmma with Vt.
// values written contiguously as [B,H,S,hd] == reshape(B, 2048, 384).
// ---------------------------------------------------------------------------
__global__ __launch_bounds__(256) void attn_kernel(
    const float* __restrict__ Q, const float* __restrict__ K,
    const float* __restrict__ Vt, float* __restrict__ vals) {
  __shared__ float lds[N_HEADS * 256];   // 8 x 16x16 fp32 tiles = 8 KB

  const int lane  = threadIdx.x;
  const int h     = threadIdx.y;
  const int lmod  = lane & 15;
  const int lhalf = lane >> 4;
  const int b     = blockIdx.x >> 7;
  const int qbase = (blockIdx.x & 127) * 16;
  const size_t bh = (size_t)(b * N_HEADS + h);

  // preload Q fragments, folding in 1/sqrt(48)
  const float scale = 0.14433756729740643f;
  const float* qrow = Q + (bh * SEQ + qbase + lmod) * HEAD_DIM + 2 * lhalf;
  v2f qf[12];
#pragma unroll
  for (int c = 0; c < 12; ++c) {
    const float2 t = *reinterpret_cast<const float2*>(qrow + 4 * c);
    qf[c] = mkv2(t.x * scale, t.y * scale);
  }

  v8f o0 = {}, o1 = {}, o2 = {};
  float* myTile = lds + h * 256;
  // A-frag read offset: element (M=lmod, N=kk) lives at (M%8)*32+(M/8)*16+N
  const int aoff = h * 256 + (lmod & 7) * 32 + (lmod >> 3) * 16 + 2 * lhalf;

  for (int kt = 0; kt < SEQ / 16; ++kt) {
    const int kbase = kt * 16;

    // scores(16q x 16k) for head h
    v8f s = {};
    const float* krow = K + (bh * SEQ + kbase + lmod) * HEAD_DIM + 2 * lhalf;
#pragma unroll
    for (int c = 0; c < 12; ++c) {
      const float2 t = *reinterpret_cast<const float2*>(krow + 4 * c);
      s = wmma_f32(qf[c], mkv2(t.x, t.y), s);
    }
#pragma unroll
    for (int r = 0; r < 8; ++r) myTile[r * 32 + lane] = s[r];
    __syncthreads();

    // softmax over heads: wave h normalizes D-layout row r==h of every head
    {
      float v[N_HEADS];
#pragma unroll
      for (int hh = 0; hh < N_HEADS; ++hh) v[hh] = lds[hh * 256 + h * 32 + lane];
      float m = v[0];
#pragma unroll
      for (int hh = 1; hh < N_HEADS; ++hh) m = fmaxf(m, v[hh]);
      float sum = 0.0f;
#pragma unroll
      for (int hh = 0; hh < N_HEADS; ++hh) { v[hh] = __expf(v[hh] - m); sum += v[hh]; }
      const float inv = 1.0f / sum;
#pragma unroll
      for (int hh = 0; hh < N_HEADS; ++hh) lds[hh * 256 + h * 32 + lane] = v[hh] * inv;
    }
    __syncthreads();

    // reload attn as A-fragments (LDS does the D->A transpose)
    v2f af[4];
#pragma unroll
    for (int c = 0; c < 4; ++c) {
      const float2 t = *reinterpret_cast<const float2*>(&lds[aoff + 4 * c]);
      af[c] = mkv2(t.x, t.y);
    }
    // attn(16q x 16k) @ V(16k x 48d); Vt gives k contiguous for float2 loads
    const float* vtbase = Vt + (bh * HEAD_DIM + lmod) * SEQ + kbase + 2 * lhalf;
#pragma unroll
    for (int c = 0; c < 4; ++c) {
      const float2 t0 = *reinterpret_cast<const float2*>(vtbase + 4 * c);
      const float2 t1 = *reinterpret_cast<const float2*>(vtbase + 16 * (size_t)SEQ + 4 * c);
      const float2 t2 = *reinterpret_cast<const float2*>(vtbase + 32 * (size_t)SEQ + 4 * c);
      o0 = wmma_f32(af[c], mkv2(t0.x, t0.y), o0);
      o1 = wmma_f32(af[c], mkv2(t1.x, t1.y), o1);
      o2 = wmma_f32(af[c], mkv2(t2.x, t2.y), o2);
    }
    // barrier #1 of the next iteration orders these LDS reads vs. next writes
  }

  // values[b,h,q,d], contiguous == reference reshape to (B, 2048, 384)
#pragma unroll
  for (int r = 0; r < 8; ++r) {
    const int q = r + 8 * lhalf;
    float* row = vals + (bh * SEQ + qbase + q) * HEAD_DIM;
    row[lmod]      = o0[r];
    row[16 + lmod] = o1[r];
    row[32 + lmod] = o2[r];
  }
}

// ---------------------------------------------------------------------------
// Kernel C: out[m,e] = vals[m,:] . W_o[e,:] + b_o[e]; 16x64 tile per wave.
// ---------------------------------------------------------------------------
__global__ __launch_bounds__(256) void out_kernel(
    const float* __restrict__ vals, const float* __restrict__ Wo,
    const float* __restrict__ bo, float* __restrict__ out) {
  const int lane  = threadIdx.x;
  const int lmod  = lane & 15;
  const int lhalf = lane >> 4;
  const int tile  = blockIdx.x * 8 + threadIdx.y;      // 16x64 tiles
  const int mbase = (tile / (D_MODEL / 64)) * 16;
  const int ebase = (tile % (D_MODEL / 64)) * 64;

  const float* arow = vals + (size_t)(mbase + lmod) * D_MODEL + 2 * lhalf;
  const float* brow0 = Wo + (size_t)(ebase +  0 + lmod) * D_MODEL + 2 * lhalf;
  const float* brow1 = Wo + (size_t)(ebase + 16 + lmod) * D_MODEL + 2 * lhalf;
  const float* brow2 = Wo + (size_t)(ebase + 32 + lmod) * D_MODEL + 2 * lhalf;
  const float* brow3 = Wo + (size_t)(ebase + 48 + lmod) * D_MODEL + 2 * lhalf;

  v8f acc0 = {}, acc1 = {}, acc2 = {}, acc3 = {};
#pragma unroll 4
  for (int c = 0; c < D_MODEL / 4; ++c) {
    const float2 af = *reinterpret_cast<const float2*>(arow  + 4 * c);
    const float2 b0 = *reinterpret_cast<const float2*>(brow0 + 4 * c);
    const float2 b1 = *reinterpret_cast<const float2*>(brow1 + 4 * c);
    const float2 b2 = *reinterpret_cast<const float2*>(brow2 + 4 * c);
    const float2 b3 = *reinterpret_cast<const float2*>(brow3 + 4 * c);
    const v2f av = mkv2(af.x, af.y);
    acc0 = wmma_f32(av, mkv2(b0.x, b0.y), acc0);
    acc1 = wmma_f32(av, mkv2(b1.x, b1.y), acc1);
    acc2 = wmma_f32(av, mkv2(b2.x, b2.y), acc2);
    acc3 = wmma_f32(av, mkv2(b3.x, b3.y), acc3);
  }

  v8f accs[4] = {acc0, acc1, acc2, acc3};
#pragma unroll
  for (int j = 0; j < 4; ++j) {
    const float bias = bo[ebase + 16 * j + lmod];
#pragma unroll
    for (int r = 0; r < 8; ++r) {
      const int m = mbase + r + 8 * lhalf;
      out[(size_t)m * D_MODEL + ebase + 16 * j + lmod] = accs[j][r] + bias;
    }
  }
}

// ---------------------------------------------------------------------------
extern "C" void kernel_launch(void* const* d_in, const int* in_sizes, int n_in,
                              void* d_out, int out_size, void* d_ws, size_t ws_size,
                              hipStream_t stream) {
  const float* x    = (const float*)d_in[0];
  const float* Wqkv = (const float*)d_in[1];
  const float* bqkv = (const float*)d_in[2];
  const float* Wo   = (const float*)d_in[3];
  const float* bo   = (const float*)d_in[4];
  float* out = (float*)d_out;

  // workspace layout (floats): Q | K | Vt | vals, each B*H*S*hd = 6291456
  float* ws   = (float*)d_ws;
  float* Q    = ws;
  float* K    = ws + 6291456;
  float* Vt   = ws + 12582912;
  float* vals = ws + 18874368;   // total 96 MB

  dim3 blk(32, 8);  // 8 wave32 waves
  // 16x64 tiles: (16384/16)*(1152/64) = 18432 tiles / 8 waves = 2304 blocks
  qkv_kernel<<<(M_TOTAL / 16) * (THREE_D / 64) / 8, blk, 0, stream>>>(x, Wqkv, bqkv, Q, K, Vt);
  attn_kernel<<<BATCH * (SEQ / 16), blk, 0, stream>>>(Q, K, Vt, vals);
  // (16384/16)*(384/64) = 6144 tiles / 8 waves = 768 blocks
  out_kernel<<<(M_TOTAL / 16) * (D_MODEL / 64) / 8, blk, 0, stream>>>(vals, Wo, bo, out);
}